// SphericalHarmonicsGaussianKernels_65309272703466
// MI455X (gfx1250) — compile-verified
//
#include <hip/hip_runtime.h>
#include <hip/hip_bf16.h>

// ---------------------------------------------------------------------------
// SphericalHarmonicsGaussianKernels for MI455X (gfx1250, wave32).
//
// Strategy (memory-bound: 128 MiB output write dominates @23.3 TB/s):
//  Pass 1: recompute sh = Y*monoms per point with V_WMMA_F32_16X16X4_F32,
//          reduce degree-grouped L2 per (b,v), sqrt, deterministic partials.
//  Pass 2: recompute sh again (cheap) and stream the 128 MiB output with
//          float4 stores, folding shells and 1/norm.
//  Per-point math uses native v_rcp_f32 / v_sqrt_f32 (1-ulp) instead of the
//  IEEE division/sqrt fixup chains — this is hot per-point code.
// ---------------------------------------------------------------------------

typedef float v2f __attribute__((ext_vector_type(2)));
typedef float v8f __attribute__((ext_vector_type(8)));

#define NB      8
#define NV      2048
#define NP      32
#define NMON    20
#define NSH     16
#define GAUSS   16.0f

#define WAVES        8
#define BLOCKS_PER_B 16
#define V_PER_BLOCK  (NV / BLOCKS_PER_B)    // 128
#define V_PER_WAVE   (V_PER_BLOCK / WAVES)  // 16
#define WAVES_PER_B  (BLOCKS_PER_B * WAVES) // 128

__device__ __forceinline__ float fastRcp(float x)  { return __builtin_amdgcn_rcpf(x); }
__device__ __forceinline__ float fastSqrt(float x) { return __builtin_amdgcn_sqrtf(x); }

// monomial order = sorted lexicographic (i,j,k), value = nx^i * ny^j * nz^k
__device__ __forceinline__ void pointEval(float x, float y, float z,
                                          float* mono, float& dist) {
  float r2 = x * x + y * y + z * z;
  dist = fastSqrt(r2);
  float inv = fastRcp(fmaxf(dist, 1e-6f));
  float nx = -x * inv, ny = -y * inv, nz = -z * inv;
  float nx2 = nx * nx, ny2 = ny * ny, nz2 = nz * nz;
  mono[0] = 1.0f;      mono[1] = nz;        mono[2] = nz2;      mono[3] = nz2 * nz;
  mono[4] = ny;        mono[5] = ny * nz;   mono[6] = ny * nz2;
  mono[7] = ny2;       mono[8] = ny2 * nz;  mono[9] = ny2 * ny;
  mono[10] = nx;       mono[11] = nx * nz;  mono[12] = nx * nz2;
  mono[13] = nx * ny;  mono[14] = nx * ny * nz; mono[15] = nx * ny2;
  mono[16] = nx2;      mono[17] = nx2 * nz; mono[18] = nx2 * ny; mono[19] = nx2 * nx;
}

__device__ __forceinline__ void shellEval(float dist, float* s4) {
  float d0 = dist;
  float d1 = dist - (1.0f / 3.0f);
  float d2 = dist - (2.0f / 3.0f);
  float d3 = dist - 1.0f;
  float e0 = __expf(-GAUSS * d0 * d0);
  float e1 = __expf(-GAUSS * d1 * d1);
  float e2 = __expf(-GAUSS * d2 * d2);
  float e3 = __expf(-GAUSS * d3 * d3);
  float sum = e0 + e1 + e2 + e3;
  float invs = fastRcp(fmaxf(sum, 1e-6f));
  float gate = (dist <= 1.0f) ? invs : 0.0f;
  s4[0] = e0 * gate; s4[1] = e1 * gate; s4[2] = e2 * gate; s4[3] = e3 * gate;
}

// Load Y into A-fragment registers: 16x4 f32 A layout per chunk c (K=4c..4c+3):
//   VGPR0: lanes0-15 K=4c+0, lanes16-31 K=4c+2 ; VGPR1: K=4c+1 / K=4c+3
__device__ __forceinline__ void loadYFrags(const float* __restrict__ Y,
                                           int m16, int half, v2f* av) {
#pragma unroll
  for (int c = 0; c < 5; ++c) {
    const float* yp = Y + m16 * NMON + 4 * c + 2 * half;
    av[c].x = yp[0];
    av[c].y = yp[1];
  }
}

// 5 chained V_WMMA_F32_16X16X4_F32: D(16 sh x 16 pts) = Y(16x20) * monoms^T(20x16)
__device__ __forceinline__ v8f shTile(const float (*s_mono)[NMON],
                                      int tile, int m16, int half, const v2f* av) {
  v8f d = {0.f, 0.f, 0.f, 0.f, 0.f, 0.f, 0.f, 0.f};
#pragma unroll
  for (int c = 0; c < 5; ++c) {
    v2f bb;
    const float* mp = &s_mono[tile * 16 + m16][4 * c + 2 * half];
    bb.x = mp[0];
    bb.y = mp[1];
    d = __builtin_amdgcn_wmma_f32_16x16x4_f32(false, av[c], false, bb,
                                              (short)0, d, false, false);
  }
  return d;
}

// -------------------- Pass 1: per-(b,v) norm reduction ----------------------
__global__ __launch_bounds__(256) void k_reduce(const float* __restrict__ patches,
                                                const float* __restrict__ Y,
                                                float* __restrict__ part) {
  __shared__ float s_mono[WAVES][NP][NMON];
  const int tid = threadIdx.x;
  const int wave = tid >> 5;
  const int lane = tid & 31;
  const int half = lane >> 4;
  const int m16 = lane & 15;
  const int b = blockIdx.y;

  v2f av[5];
  loadYFrags(Y, m16, half, av);

  float vsum[16];
#pragma unroll
  for (int j = 0; j < 16; ++j) vsum[j] = 0.0f;

  for (int i = 0; i < V_PER_WAVE; ++i) {
    const int v = blockIdx.x * V_PER_BLOCK + i * WAVES + wave;
    const float* pp = patches + (((size_t)b * NV + v) * NP + lane) * 3;
    float x = pp[0], y = pp[1], z = pp[2];
    float mono[NMON], dist;
    pointEval(x, y, z, mono, dist);
#pragma unroll
    for (int k = 0; k < NMON; ++k) s_mono[wave][lane][k] = mono[k];

    // degree-group accumulators (compile-time indexed; mapped by lane-half)
    float accA[4] = {0, 0, 0, 0};   // row r=0   (m=0 | m=8)
    float accB[4] = {0, 0, 0, 0};   // rows 1-3  (m=1-3 | m=9-11)
    float accC[4] = {0, 0, 0, 0};   // rows 4-7  (m=4-7 | m=12-15)

#pragma unroll
    for (int t = 0; t < 2; ++t) {
      v8f d = shTile(s_mono[wave], t, m16, half, av);
      float dq = __shfl(dist, t * 16 + m16, 32);  // dist of this column's point
      float s4[4];
      shellEval(dq, s4);
#pragma unroll
      for (int r = 0; r < 8; ++r) {
#pragma unroll
        for (int s = 0; s < 4; ++s) {
          float tv = d[r] * s4[s];
          float sq = tv * tv;
          if (r == 0)      accA[s] += sq;
          else if (r < 4)  accB[s] += sq;
          else             accC[s] += sq;
        }
      }
    }

    // fold to (degree, shell): deg0=m0, deg1=m1-3, deg2=m4-8, deg3=m9-15
    float tot[16];
#pragma unroll
    for (int s = 0; s < 4; ++s) {
      tot[0 * 4 + s] = half ? 0.0f : accA[s];
      tot[1 * 4 + s] = half ? 0.0f : accB[s];
      tot[2 * 4 + s] = half ? accA[s] : accC[s];
      tot[3 * 4 + s] = half ? (accB[s] + accC[s]) : 0.0f;
    }
#pragma unroll
    for (int off = 16; off >= 1; off >>= 1) {
#pragma unroll
      for (int j = 0; j < 16; ++j) tot[j] += __shfl_xor(tot[j], off, 32);
    }
#pragma unroll
    for (int j = 0; j < 16; ++j) vsum[j] += fastSqrt(tot[j]);
  }

  // deterministic partials: part[b][waveSeq][16]
  if (lane == 0) {
    const int wseq = blockIdx.x * WAVES + wave;
    float* dst = part + ((size_t)b * WAVES_PER_B + wseq) * 16;
#pragma unroll
    for (int j = 0; j < 16; ++j) dst[j] = vsum[j];
  }
}

// -------------------- Pass 2: normalize + stream output ---------------------
__global__ __launch_bounds__(256) void k_output(const float* __restrict__ patches,
                                                const float* __restrict__ Y,
                                                const float* __restrict__ part,
                                                float* __restrict__ out) {
  __shared__ float s_mono[WAVES][NP][NMON];
  __shared__ float4 s_inv[NSH];
  const int tid = threadIdx.x;
  const int wave = tid >> 5;
  const int lane = tid & 31;
  const int half = lane >> 4;
  const int m16 = lane & 15;
  const int b = blockIdx.y;

  if (tid < 64) {  // 16 sh rows x 4 shells; precise math (runs once per block)
    const int m = tid >> 2, s = tid & 3;
    const int deg = (m == 0) ? 0 : (m < 4) ? 1 : (m < 9) ? 2 : 3;
    const float* src = part + (size_t)b * WAVES_PER_B * 16 + deg * 4 + s;
    float sum = 0.0f;
    for (int w = 0; w < WAVES_PER_B; ++w) sum += src[(size_t)w * 16];
    float ml = fmaxf(sum * (1.0f / NV), 1e-8f);
    ((float*)s_inv)[tid] = 1.0f / ml;
  }
  __syncthreads();

  float4 invr[8];
#pragma unroll
  for (int r = 0; r < 8; ++r) invr[r] = s_inv[half * 8 + r];

  v2f av[5];
  loadYFrags(Y, m16, half, av);

  for (int i = 0; i < V_PER_WAVE; ++i) {
    const int v = blockIdx.x * V_PER_BLOCK + i * WAVES + wave;
    const float* pp = patches + (((size_t)b * NV + v) * NP + lane) * 3;
    float x = pp[0], y = pp[1], z = pp[2];
    float mono[NMON], dist;
    pointEval(x, y, z, mono, dist);
#pragma unroll
    for (int k = 0; k < NMON; ++k) s_mono[wave][lane][k] = mono[k];

#pragma unroll
    for (int t = 0; t < 2; ++t) {
      v8f d = shTile(s_mono[wave], t, m16, half, av);
      float dq = __shfl(dist, t * 16 + m16, 32);
      float s4[4];
      shellEval(dq, s4);
      // point q's 64-float block; lane-half covers rows m=half*8 .. half*8+7
      float* obase = out + ((((size_t)b * NV + v) * NP) + t * 16 + m16) * 64
                         + (size_t)half * 32;
#pragma unroll
      for (int r = 0; r < 8; ++r) {
        float4 o;
        o.x = d[r] * s4[0] * invr[r].x;
        o.y = d[r] * s4[1] * invr[r].y;
        o.z = d[r] * s4[2] * invr[r].z;
        o.w = d[r] * s4[3] * invr[r].w;
        *(float4*)(obase + r * 4) = o;  // global_store_b128
      }
    }
  }
}

extern "C" void kernel_launch(void* const* d_in, const int* in_sizes, int n_in,
                              void* d_out, int out_size, void* d_ws, size_t ws_size,
                              hipStream_t stream) {
  const float* patches = (const float*)d_in[0];  // (8,2048,32,3) f32
  const float* Y       = (const float*)d_in[1];  // (16,20) f32
  float* out = (float*)d_out;                    // (8,2048,32,16,4) f32
  float* part = (float*)d_ws;                    // 8*128*16 f32 = 64 KiB partials

  dim3 grid(BLOCKS_PER_B, NB);
  dim3 block(256);
  k_reduce<<<grid, block, 0, stream>>>(patches, Y, part);
  k_output<<<grid, block, 0, stream>>>(patches, Y, part, out);
}